// GCN_9405978378565
// MI455X (gfx1250) — compile-verified
//
#include <hip/hip_runtime.h>

typedef __attribute__((ext_vector_type(2))) float v2f;
typedef __attribute__((ext_vector_type(8))) float v8f;

#define NFEAT 128
#define HID   128
#define EMB   64

// ---------------- degree / symmetric norm ----------------
__global__ void k_deg_init(float* deg, int n) {
  int i = blockIdx.x * blockDim.x + threadIdx.x;
  if (i < n) deg[i] = 1.0f;                       // self-loop contributes 1
}
__global__ void k_deg_edges(const int* __restrict__ dst, float* deg, int e) {
  int i = blockIdx.x * blockDim.x + threadIdx.x;
  if (i < e) atomicAdd(&deg[dst[i]], 1.0f);
}
__global__ void k_deg_rsqrt(float* deg, int n) {
  int i = blockIdx.x * blockDim.x + threadIdx.x;
  if (i < n) deg[i] = rsqrtf(deg[i]);             // deg >= 1 always
}

// ---------------- GEMM: C[n x FOUT] = A[n x FIN] @ W[FIN x FOUT] ----------------
// One wave computes one 16x16 C tile via V_WMMA_F32_16X16X4_F32 (K stepped by 4).
// Block = 256 threads = 8 waves = 128 rows x 16 cols. W column-slice staged in LDS.
// Tail rows: A address is clamped to row n-1 and loaded UNCONDITIONALLY — a bad A
// row M only pollutes C row M, and C rows >= n are never stored. Stores use a
// tile-level guard so the common path is 8 unconditional b32 stores at immediate
// offsets from one base (FOUT is compile-time).
template <int FIN, int FOUT>
__global__ __launch_bounds__(256) void k_gemm_wmma(
    const float* __restrict__ A, const float* __restrict__ W,
    float* __restrict__ C, int n) {
  __shared__ float Wlds[FIN * 16];                // 8 KB for FIN=128
  const int colBase = blockIdx.y * 16;
  for (int i = threadIdx.x; i < FIN * 16; i += 256) {
    int k = i >> 4, c = i & 15;
    Wlds[i] = W[k * FOUT + colBase + c];
  }
  __syncthreads();

  const int wave = threadIdx.x >> 5;              // wave32
  const int lane = threadIdx.x & 31;
  const int half = lane >> 4;                     // lanes 16..31 carry K+2,K+3
  const int lr   = lane & 15;
  const int rowBase = blockIdx.x * 128 + wave * 16;
  int arow = rowBase + lr;
  if (arow >= n) arow = n - 1;                    // clamp: safe, result discarded
  const float2* __restrict__ Ap2 =
      (const float2*)(A + (size_t)arow * FIN);    // 8B-aligned pairs

  v8f acc = {};
#pragma unroll 8
  for (int k = 0; k < FIN; k += 4) {
    const int ka = k + half * 2;
    float2 av = Ap2[(k >> 1) + half];             // single global_load_b64
    v2f a, b;
    a.x = av.x;                                   // A 16x4: row=lane%16, K pair per half
    a.y = av.y;
    b.x = Wlds[ka * 16 + lr];                     // B 4x16: col=lane%16 (ds_load_2addr)
    b.y = Wlds[(ka + 1) * 16 + lr];
    acc = __builtin_amdgcn_wmma_f32_16x16x4_f32(false, a, false, b,
                                                (short)0, acc, false, false);
  }

  const int row0 = rowBase + half * 8;            // C: vgpr r -> M = r (+8 for hi half)
  const int col  = colBase + lr;
  if (rowBase + 16 <= n) {                        // full tile: unconditional stores
    float* __restrict__ Cp = C + (size_t)row0 * FOUT + col;
#pragma unroll
    for (int r = 0; r < 8; ++r) Cp[r * FOUT] = acc[r];  // immediate offsets
  } else if (rowBase < n) {                       // partial tile (n % 16 != 0 only)
#pragma unroll
    for (int r = 0; r < 8; ++r) {
      int row = row0 + r;
      if (row < n) C[(size_t)row * FOUT + col] = acc[r];
    }
  }
}

// ---------------- aggregation: out = dis[i]^2 * h[i] + bias  (self-loop term) ----
template <int F>
__global__ void k_agg_init(const float* __restrict__ h, const float* __restrict__ dis,
                           const float* __restrict__ bias, float* __restrict__ out, int n) {
  int idx = blockIdx.x * blockDim.x + threadIdx.x;
  if (idx >= n * F) return;
  int i = idx / F, f = idx % F;
  float w = dis[i];
  out[idx] = w * w * h[idx] + bias[f];
}

// ---------------- edge scatter: out[dst] += dis[src]*dis[dst] * h[src] ----------
// One wave per edge; lanes stride the feature dim -> coalesced loads & atomics.
// Working set (h, out: 25.6 MB each) fits in the 192 MB L2, so the random
// gathers and f32 atomic scatters resolve at L2 bandwidth, not HBM.
template <int F>
__global__ void k_edge_scatter(const float* __restrict__ h, const int* __restrict__ src,
                               const int* __restrict__ dst, const float* __restrict__ dis,
                               float* __restrict__ out, int e) {
  int gid  = blockIdx.x * blockDim.x + threadIdx.x;
  int edge = gid >> 5;
  int lane = gid & 31;
  if (edge >= e) return;
  int s = src[edge], d = dst[edge];
  float w = dis[s] * dis[d];
  const float* hs = h + (size_t)s * F;
  float* od = out + (size_t)d * F;
#pragma unroll
  for (int i = 0; i < F / 32; ++i) {
    int f = lane + i * 32;
    atomicAdd(&od[f], w * hs[f]);                 // global_atomic_add_f32, no return
  }
}

__global__ void k_relu(float* x, int count) {
  int i = blockIdx.x * blockDim.x + threadIdx.x;
  if (i < count) x[i] = fmaxf(x[i], 0.0f);
}

// ---------------- final projection: out[i] = h3[i,:] . Wout + bout ---------------
__global__ void k_final(const float* __restrict__ h3, const float* __restrict__ Wout,
                        const float* __restrict__ bout, float* __restrict__ out, int n) {
  int i = blockIdx.x * blockDim.x + threadIdx.x;
  if (i >= n) return;
  float acc = bout[0];
  const float* hp = h3 + (size_t)i * EMB;
#pragma unroll
  for (int f = 0; f < EMB; ++f) acc += hp[f] * Wout[f];
  out[i] = acc;
}

extern "C" void kernel_launch(void* const* d_in, const int* in_sizes, int n_in,
                              void* d_out, int out_size, void* d_ws, size_t ws_size,
                              hipStream_t stream) {
  const float* x    = (const float*)d_in[0];
  const int*   ei   = (const int*)d_in[1];
  const float* W1   = (const float*)d_in[2];
  const float* b1   = (const float*)d_in[3];
  const float* Wh   = (const float*)d_in[4];
  const float* bh   = (const float*)d_in[5];
  const float* W2   = (const float*)d_in[6];
  const float* b2   = (const float*)d_in[7];
  const float* Wout = (const float*)d_in[8];
  const float* bout = (const float*)d_in[9];

  const int N = in_sizes[0] / NFEAT;              // 50000
  const int E = in_sizes[1] / 2;                  // 800000
  const int* src = ei;                            // edge_index[0]
  const int* dst = ei + E;                        // edge_index[1]

  float* ws   = (float*)d_ws;
  float* dis  = ws;                               // N floats (deg -> rsqrt in place)
  float* buf0 = ws + ((N + 63) & ~63);            // N x 128
  float* buf1 = buf0 + (size_t)N * HID;           // N x 128
  float* outp = (float*)d_out;                    // [0,N): out
  float* h3   = outp + N;                         // [N, N+N*64): h (layer-3 output)

  const int T = 256;
  dim3 gN((N + T - 1) / T), gE((E + T - 1) / T);
  dim3 gNF128((N * HID + T - 1) / T), gNF64((N * EMB + T - 1) / T);
  dim3 gEdgeWave(((E * 32) + T - 1) / T);
  dim3 gGemm((N + 127) / 128, HID / 16);
  dim3 gGemm64((N + 127) / 128, EMB / 16);

  // norm
  k_deg_init<<<gN, T, 0, stream>>>(dis, N);
  k_deg_edges<<<gE, T, 0, stream>>>(dst, dis, E);
  k_deg_rsqrt<<<gN, T, 0, stream>>>(dis, N);

  // layer 1: h1 = relu(Agg(x @ W1) + b1)
  k_gemm_wmma<NFEAT, HID><<<gGemm, T, 0, stream>>>(x, W1, buf0, N);
  k_agg_init<HID><<<gNF128, T, 0, stream>>>(buf0, dis, b1, buf1, N);
  k_edge_scatter<HID><<<gEdgeWave, T, 0, stream>>>(buf0, src, dst, dis, buf1, E);
  k_relu<<<gNF128, T, 0, stream>>>(buf1, N * HID);

  // layer 2: h2 = relu(Agg(h1 @ Wh) + bh)
  k_gemm_wmma<HID, HID><<<gGemm, T, 0, stream>>>(buf1, Wh, buf0, N);
  k_agg_init<HID><<<gNF128, T, 0, stream>>>(buf0, dis, bh, buf1, N);
  k_edge_scatter<HID><<<gEdgeWave, T, 0, stream>>>(buf0, src, dst, dis, buf1, E);
  k_relu<<<gNF128, T, 0, stream>>>(buf1, N * HID);

  // layer 3: h3 = Agg(h2 @ W2) + b2   (no relu), written straight into d_out
  k_gemm_wmma<HID, EMB><<<gGemm64, T, 0, stream>>>(buf1, W2, buf0, N);
  k_agg_init<EMB><<<gNF64, T, 0, stream>>>(buf0, dis, b2, h3, N);
  k_edge_scatter<EMB><<<gEdgeWave, T, 0, stream>>>(buf0, src, dst, dis, h3, E);

  // out = h3 @ Wout + bout
  k_final<<<gN, T, 0, stream>>>(h3, Wout, bout, outp, N);
}